// AttentionLayer_4475355922565
// MI455X (gfx1250) — compile-verified
//
#include <hip/hip_runtime.h>
#include <hip/hip_bf16.h>
#include <math.h>

// Problem constants (match reference)
#define GN    8192      // nodes
#define GDIN  512
#define GDOUT 256
#define GALPHA 0.2f
#define GEPS   1e-12f

#define TBL_BITS 20
#define TBL_SIZE (1u << TBL_BITS)
#define TBL_MASK (TBL_SIZE - 1u)
#define KEY_EMPTY 0xFFFFFFFFu

typedef float v8f __attribute__((ext_vector_type(8)));
typedef float v2f __attribute__((ext_vector_type(2)));

// ---------------------------------------------------------------------------
// K0: init workspace (acc, denom, sumH zero; hash keys EMPTY, values zero)
// ---------------------------------------------------------------------------
__global__ void k_init(float* __restrict__ acc, float* __restrict__ denom,
                       float* __restrict__ sumH, unsigned* __restrict__ hkeys,
                       float* __restrict__ hvals) {
    unsigned idx = blockIdx.x * blockDim.x + threadIdx.x;
    if (idx < (unsigned)(GN * GDOUT)) acc[idx] = 0.0f;
    if (idx < GN) denom[idx] = 0.0f;
    if (idx < GDOUT) sumH[idx] = 0.0f;
    if (idx < TBL_SIZE) { hkeys[idx] = KEY_EMPTY; hvals[idx] = 0.0f; }
}

// ---------------------------------------------------------------------------
// K1: h = x @ weight using V_WMMA_F32_16X16X4_F32.
// One wave computes one 16x16 tile of h; K loop = 512/4 = 128 WMMAs.
// A layout (16x4 f32): lane l<16 -> row M=l, VGPR0=K0, VGPR1=K1;
//                      lane l>=16 -> row M=l-16, VGPR0=K2, VGPR1=K3.
// B layout (4x16 f32): VGPR0 lanes0-15 = row K0 (N=lane), lanes16-31 = row K2;
//                      VGPR1 = rows K1 / K3.
// C/D layout: VGPR v -> M=v (lanes0-15, N=lane) / M=v+8 (lanes16-31).
// ---------------------------------------------------------------------------
__global__ void k_gemm_wmma(const float* __restrict__ x,
                            const float* __restrict__ w,
                            float* __restrict__ h) {
    const int lane  = threadIdx.x & 31;
    const int waveId = threadIdx.x >> 5;
    const int tile  = blockIdx.x * 8 + waveId;        // 8192 tiles total
    const int mTile = tile >> 4;                      // 512 M tiles
    const int nTile = tile & 15;                      // 16 N tiles
    const int m0 = mTile * 16;
    const int n0 = nTile * 16;

    const int row   = lane & 15;
    const int khalf = lane >> 4;                      // 0: K0/K1, 1: K2/K3

    const float* arow = x + (size_t)(m0 + row) * GDIN + 2 * khalf;
    const int bcol = n0 + row;

    v8f acc = {};
#pragma unroll 4
    for (int k0 = 0; k0 < GDIN; k0 += 4) {
        const int kk = k0 + 2 * khalf;
        v2f a = *(const v2f*)(arow + k0);             // x[m0+row][kk], [kk+1]
        v2f b;
        b.x = w[(size_t)kk * GDOUT + bcol];           // weight[kk][bcol]
        b.y = w[(size_t)(kk + 1) * GDOUT + bcol];     // weight[kk+1][bcol]
        acc = __builtin_amdgcn_wmma_f32_16x16x4_f32(
            /*neg_a=*/false, a, /*neg_b=*/false, b,
            /*c_mod=*/(short)0, acc, /*reuse_a=*/false, /*reuse_b=*/false);
    }

#pragma unroll
    for (int v = 0; v < 8; ++v) {
        const int M = v + 8 * khalf;
        h[(size_t)(m0 + M) * GDOUT + n0 + row] = acc[v];
    }
}

// ---------------------------------------------------------------------------
// K2: column sums of h -> sumH[d] (64 blocks, each handles 128 rows)
// ---------------------------------------------------------------------------
__global__ void k_colsum(const float* __restrict__ h, float* __restrict__ sumH) {
    const int d = threadIdx.x;                        // 0..255
    const int r0 = blockIdx.x * 128;
    float s = 0.0f;
    for (int r = 0; r < 128; ++r)
        s += h[(size_t)(r0 + r) * GDOUT + d];
    atomicAdd(&sumH[d], s);
}

// ---------------------------------------------------------------------------
// K3: Wh1[i] = h[i]·a[0:256], Wh2[i] = h[i]·a[256:512]. One wave per row.
// ---------------------------------------------------------------------------
__global__ void k_rowdot(const float* __restrict__ h, const float* __restrict__ a,
                         float* __restrict__ Wh1, float* __restrict__ Wh2) {
    const int lane = threadIdx.x & 31;
    const int wave = (blockIdx.x * blockDim.x + threadIdx.x) >> 5;
    const float* hr = h + (size_t)wave * GDOUT + lane * 8;
    float s1 = 0.0f, s2 = 0.0f;
#pragma unroll
    for (int j = 0; j < 8; ++j) {
        float hv = hr[j];
        s1 += hv * a[lane * 8 + j];
        s2 += hv * a[GDOUT + lane * 8 + j];
    }
#pragma unroll
    for (int m = 16; m >= 1; m >>= 1) {
        s1 += __shfl_xor(s1, m, 32);
        s2 += __shfl_xor(s2, m, 32);
    }
    if (lane == 0) { Wh1[wave] = s1; Wh2[wave] = s2; }
}

// ---------------------------------------------------------------------------
// K4: scatter edges into hash table. value accumulates duplicate sum exactly,
// matching `.at[row,col].add(e[row,col])` semantics.
// ---------------------------------------------------------------------------
__global__ void k_edge_scatter(const long long* __restrict__ ei, int E,
                               const float* __restrict__ Wh1,
                               const float* __restrict__ Wh2,
                               unsigned* __restrict__ hkeys,
                               float* __restrict__ hvals) {
    const int e = blockIdx.x * blockDim.x + threadIdx.x;
    if (e >= E) return;
    const int i = (int)ei[e];
    const int c = (int)ei[E + e];
    float s = Wh1[i] + Wh2[c];
    float v = (s > 0.0f) ? s : GALPHA * s;            // leaky_relu
    unsigned key = ((unsigned)i << 13) | (unsigned)c; // 26-bit key
    unsigned slot = (key * 2654435761u) & TBL_MASK;
    for (;;) {
        unsigned prev = atomicCAS(&hkeys[slot], KEY_EMPTY, key);
        if (prev == KEY_EMPTY || prev == key) {
            atomicAdd(&hvals[slot], v);
            break;
        }
        slot = (slot + 1u) & TBL_MASK;
    }
}

// ---------------------------------------------------------------------------
// K5: per occupied slot: w = exp(val)-1; denom[i] += w; acc[i][:] += w*h[c][:]
// One wave per slot; each lane covers 8 output dims.
// ---------------------------------------------------------------------------
__global__ void k_slot_gather(const unsigned* __restrict__ hkeys,
                              const float* __restrict__ hvals,
                              const float* __restrict__ h,
                              float* __restrict__ denom,
                              float* __restrict__ acc) {
    const int lane = threadIdx.x & 31;
    const int waveId = threadIdx.x >> 5;
    const unsigned slot = blockIdx.x * 8u + waveId;
    const unsigned key = hkeys[slot];                 // broadcast load
    if (key == KEY_EMPTY) return;
    const float w = expf(hvals[slot]) - 1.0f;
    const int i = (int)(key >> 13);
    const int c = (int)(key & 8191u);
    if (lane == 0) atomicAdd(&denom[i], w);
    const float* hc = h + (size_t)c * GDOUT + lane * 8;
    float* ar = acc + (size_t)i * GDOUT + lane * 8;
#pragma unroll
    for (int j = 0; j < 8; ++j)
        atomicAdd(&ar[j], w * hc[j]);
}

// ---------------------------------------------------------------------------
// K6: finalize. raw = (sumH + acc[i]) / (N + denom[i]); leaky; L2-normalize
// per row; + bias. One block (256 threads) per row.
// ---------------------------------------------------------------------------
__global__ void k_finalize(const float* __restrict__ acc,
                           const float* __restrict__ denom,
                           const float* __restrict__ sumH,
                           const float* __restrict__ bias,
                           float* __restrict__ out) {
    const int i = blockIdx.x;
    const int d = threadIdx.x;
    const int lane = d & 31;
    const int waveId = d >> 5;

    const float inv = 1.0f / ((float)GN + denom[i]);
    float p = (sumH[d] + acc[(size_t)i * GDOUT + d]) * inv;
    float v = (p > 0.0f) ? p : GALPHA * p;            // leaky_relu

    float ss = v * v;
#pragma unroll
    for (int m = 16; m >= 1; m >>= 1) ss += __shfl_xor(ss, m, 32);

    __shared__ float red[8];
    __shared__ float normsh;
    if (lane == 0) red[waveId] = ss;
    __syncthreads();
    if (d == 0) {
        float t = 0.0f;
#pragma unroll
        for (int j = 0; j < 8; ++j) t += red[j];
        normsh = fmaxf(sqrtf(t), GEPS);
    }
    __syncthreads();

    out[(size_t)i * GDOUT + d] = v / normsh + bias[d];
}

// ---------------------------------------------------------------------------
// launch
// ---------------------------------------------------------------------------
extern "C" void kernel_launch(void* const* d_in, const int* in_sizes, int n_in,
                              void* d_out, int out_size, void* d_ws, size_t ws_size,
                              hipStream_t stream) {
    const float*     x    = (const float*)d_in[0];
    const long long* ei   = (const long long*)d_in[1];   // int64 [2, E]
    const float*     w    = (const float*)d_in[2];
    const float*     a    = (const float*)d_in[3];
    const float*     bias = (const float*)d_in[4];
    float* out = (float*)d_out;
    const int E = in_sizes[1] / 2;

    char* ws = (char*)d_ws;
    size_t off = 0;
    float* h     = (float*)(ws + off); off += (size_t)GN * GDOUT * 4;   // 8 MB
    float* acc   = (float*)(ws + off); off += (size_t)GN * GDOUT * 4;   // 8 MB
    float* Wh1   = (float*)(ws + off); off += (size_t)GN * 4;
    float* Wh2   = (float*)(ws + off); off += (size_t)GN * 4;
    float* denom = (float*)(ws + off); off += (size_t)GN * 4;
    float* sumH  = (float*)(ws + off); off += (size_t)GDOUT * 4;
    unsigned* hkeys = (unsigned*)(ws + off); off += (size_t)TBL_SIZE * 4; // 4 MB
    float*    hvals = (float*)(ws + off);    off += (size_t)TBL_SIZE * 4; // 4 MB

    // K0: init (cover max region: acc = 2M elements)
    {
        unsigned total = (unsigned)(GN * GDOUT);
        k_init<<<(total + 255) / 256, 256, 0, stream>>>(acc, denom, sumH, hkeys, hvals);
    }
    // K1: WMMA GEMM: 8192 tiles, 8 waves/block
    k_gemm_wmma<<<1024, 256, 0, stream>>>(x, w, h);
    // K2: column sums
    k_colsum<<<64, 256, 0, stream>>>(h, sumH);
    // K3: per-row attention dots (8192 waves)
    k_rowdot<<<1024, 256, 0, stream>>>(h, a, Wh1, Wh2);
    // K4: edge scatter into hash
    k_edge_scatter<<<(E + 255) / 256, 256, 0, stream>>>(ei, E, Wh1, Wh2, hkeys, hvals);
    // K5: slot gather (1M slots, 8 waves/block)
    k_slot_gather<<<TBL_SIZE / 8, 256, 0, stream>>>(hkeys, hvals, h, denom, acc);
    // K6: finalize, one block per row
    k_finalize<<<GN, GDOUT, 0, stream>>>(acc, denom, sumH, bias, out);
}